// DeployPoseModel_6390911336923
// MI455X (gfx1250) — compile-verified
//
#include <hip/hip_runtime.h>
#include <hip/hip_bf16.h>
#include <stdint.h>

// ---------------------------------------------------------------------------
// YOLOX-pose decode, select-then-decode for MI455X (gfx1250, wave32).
//   Kernel A (fused): per-batch workgroup stages cls/obj planes into the
//     320KB-per-WGP LDS with async global->LDS b128 copies, computes
//     sigmoid(cls)*sigmoid(obj) in LDS, then does exact top-1000 via
//     8-pass radix select + bitonic sort entirely out of LDS.
//   Kernel B: decodes bbox/kpt/vis only for the 1000 survivors per batch.
// ---------------------------------------------------------------------------

#define NA      33600      // 160^2 + 80^2 + 40^2
#define TOPK    1000
#define BATCH   32
#define L0_END  25600
#define L1_END  32000
#define CHUNK   8400       // NA / 4, staged per async round
#define TPB     1024

// dynamic LDS layout (bytes)
#define LDS_SCORES   0                         // NA floats      = 134400
#define LDS_OBJ      (NA * 4)                  // CHUNK floats   =  33600
#define LDS_SEL      (NA * 4 + CHUNK * 4)      // 1024 u64       =   8192
#define LDS_HIST     (LDS_SEL + 1024 * 8)      // 256 u32        =   1024
#define LDS_TOTAL    (LDS_HIST + 256 * 4)      // 177216 B  (< 320KB/WGP)

__device__ __forceinline__ float sigmoidf(float x) {
    return 1.0f / (1.0f + expf(-x));
}

// Unique monotone key: positive-float bits high, ~idx low => descending
// sort == value-desc then index-asc (jax.lax.top_k tie order), keys unique.
__device__ __forceinline__ uint64_t pack_key(float s, uint32_t idx) {
    return ((uint64_t)__float_as_uint(s) << 32) | (uint32_t)(0xFFFFFFFFu - idx);
}

__device__ __forceinline__ uint32_t lds_off(const void* p) {
    return (uint32_t)(uintptr_t)p;     // generic LDS ptr: low 32 bits = offset
}

__device__ __forceinline__ void async_ld_b128(uint32_t ldsOff, uint64_t gaddr) {
    asm volatile("global_load_async_to_lds_b128 %0, %1, off"
                 :: "v"(ldsOff), "v"(gaddr) : "memory");
}

// ---------------------------------------------------------------------------
// Kernel A: fused scores + exact top-1000.  grid = 32, block = 1024,
// dynamic LDS = LDS_TOTAL bytes.
// ---------------------------------------------------------------------------
__global__ __launch_bounds__(TPB) void topk_fused_kernel(
    const float* __restrict__ cls0, const float* __restrict__ cls1,
    const float* __restrict__ cls2,
    const float* __restrict__ obj0, const float* __restrict__ obj1,
    const float* __restrict__ obj2,
    int* __restrict__ keep)
{
    extern __shared__ char smem[];
    float*    scoresLds = (float*)(smem + LDS_SCORES);   // NA scores
    float*    objStage  = (float*)(smem + LDS_OBJ);      // CHUNK staging
    uint64_t* sel       = (uint64_t*)(smem + LDS_SEL);   // 1024 selected keys
    uint32_t* hist      = (uint32_t*)(smem + LDS_HIST);  // 256 bins

    __shared__ uint64_t sh_prefix;
    __shared__ int      sh_rem;
    __shared__ uint32_t sh_cnt;

    const int b   = blockIdx.x;
    const int tid = threadIdx.x;

    // ---- stage cls/obj via CDNA5 async copies, 4 chunks of 8400 anchors ----
    for (int c = 0; c < 4; ++c) {
        const int base = c * CHUNK;
        for (int slot = tid; slot < CHUNK / 4; slot += TPB) {
            const int g = base + slot * 4;            // 4-aligned global anchor
            const float* cp; const float* op; int hw, off;
            if (g < L0_END)      { cp = cls0; op = obj0; hw = 25600; off = g; }
            else if (g < L1_END) { cp = cls1; op = obj1; hw = 6400;  off = g - L0_END; }
            else                 { cp = cls2; op = obj2; hw = 1600;  off = g - L1_END; }
            const uint64_t gc = (uint64_t)(uintptr_t)(cp + (size_t)b * hw + off);
            const uint64_t go = (uint64_t)(uintptr_t)(op + (size_t)b * hw + off);
            async_ld_b128(lds_off(&scoresLds[g]),       gc);
            async_ld_b128(lds_off(&objStage[slot * 4]), go);
        }
        asm volatile("s_wait_asynccnt 0" ::: "memory");
        __syncthreads();
        for (int i = tid; i < CHUNK; i += TPB)
            scoresLds[base + i] = sigmoidf(scoresLds[base + i]) *
                                  sigmoidf(objStage[i]);
        __syncthreads();
    }

    // ---- 8-pass radix select over 64-bit keys, all reads from LDS ----
    if (tid == 0) { sh_prefix = 0ull; sh_rem = TOPK; }
    __syncthreads();

    for (int stage = 0; stage < 8; ++stage) {
        const int shift = 56 - stage * 8;
        for (int i = tid; i < 256; i += TPB) hist[i] = 0u;
        __syncthreads();

        const uint64_t prefix = sh_prefix;
        const uint64_t mask = (stage == 0) ? 0ull : (~0ull << (shift + 8));
        for (int i = tid; i < NA; i += TPB) {
            const uint64_t key = pack_key(scoresLds[i], (uint32_t)i);
            if ((key & mask) == prefix)
                atomicAdd(&hist[(uint32_t)(key >> shift) & 0xFFu], 1u);
        }
        __syncthreads();

        if (tid == 0) {
            const int rem = sh_rem;
            int cum = 0;
            int d = 255;
            for (; d > 0; --d) {
                const int c = (int)hist[d];
                if (cum + c >= rem) break;
                cum += c;
            }
            sh_prefix = prefix | ((uint64_t)(uint32_t)d << shift);
            sh_rem    = rem - cum;
        }
        __syncthreads();
    }

    // ---- compact: exactly TOPK unique keys >= threshold ----
    const uint64_t thresh = sh_prefix;
    if (tid == 0) sh_cnt = 0u;
    __syncthreads();
    for (int i = tid; i < NA; i += TPB) {
        const uint64_t key = pack_key(scoresLds[i], (uint32_t)i);
        if (key >= thresh) {
            const uint32_t p = atomicAdd(&sh_cnt, 1u);
            if (p < 1024u) sel[p] = key;
        }
    }
    __syncthreads();
    if ((uint32_t)tid >= sh_cnt) sel[tid] = 0ull;    // pad to 1024
    __syncthreads();

    // ---- 1024-wide bitonic sort, descending ----
    for (int k = 2; k <= 1024; k <<= 1) {
        for (int j = k >> 1; j > 0; j >>= 1) {
            const int ixj = tid ^ j;
            if (ixj > tid) {
                const uint64_t a = sel[tid];
                const uint64_t c = sel[ixj];
                const bool desc = ((tid & k) == 0);
                if (desc ? (a < c) : (a > c)) { sel[tid] = c; sel[ixj] = a; }
            }
            __syncthreads();
        }
    }

    if (tid < TOPK)
        keep[b * TOPK + tid] =
            (int)(0xFFFFFFFFu - (uint32_t)(sel[tid] & 0xFFFFFFFFull));
}

// ---------------------------------------------------------------------------
// Kernel B: decode survivors only. One thread per (batch, rank).
// dets (B,K,5) at out[0..159999]; kpts (B,K,17,3) after.
// ---------------------------------------------------------------------------
__global__ __launch_bounds__(128) void gather_decode_kernel(
    const float* __restrict__ cls0, const float* __restrict__ cls1,
    const float* __restrict__ cls2,
    const float* __restrict__ bb0,  const float* __restrict__ bb1,
    const float* __restrict__ bb2,
    const float* __restrict__ ob0,  const float* __restrict__ ob1,
    const float* __restrict__ ob2,
    const float* __restrict__ kp0,  const float* __restrict__ kp1,
    const float* __restrict__ kp2,
    const float* __restrict__ vs0,  const float* __restrict__ vs1,
    const float* __restrict__ vs2,
    const int* __restrict__ keep, float* __restrict__ out)
{
    const int t = blockIdx.x * blockDim.x + threadIdx.x;
    if (t >= BATCH * TOPK) return;
    const int b = t / TOPK;
    const int a = keep[t];

    const float *cP, *bP, *oP, *kP, *vP;
    int w, s, pos;
    if (a < L0_END)      { cP=cls0; bP=bb0; oP=ob0; kP=kp0; vP=vs0; w=160; s=8;  pos=a; }
    else if (a < L1_END) { cP=cls1; bP=bb1; oP=ob1; kP=kp1; vP=vs1; w=80;  s=16; pos=a-L0_END; }
    else                 { cP=cls2; bP=bb2; oP=ob2; kP=kp2; vP=vs2; w=40;  s=32; pos=a-L1_END; }

    const size_t hw = (size_t)w * w;
    const float fs = (float)s;
    const float px = (float)(pos % w) * fs;
    const float py = (float)(pos / w) * fs;

    const float* kp = kP + (size_t)b * 34 * hw + pos;
    const float* vp = vP + (size_t)b * 17 * hw + pos;
    __builtin_prefetch(kp, 0, 0);                   // global_prefetch_b8
    __builtin_prefetch(vp, 0, 0);

    const float score = sigmoidf(cP[(size_t)b * hw + pos]) *
                        sigmoidf(oP[(size_t)b * hw + pos]);

    const float* bb = bP + (size_t)b * 4 * hw + pos;
    const float xc = bb[0]          * fs + px;
    const float yc = bb[hw]         * fs + py;
    const float hx = expf(bb[2*hw]) * fs * 0.5f;
    const float hy = expf(bb[3*hw]) * fs * 0.5f;

    float* det = out + (size_t)t * 5;
    det[0] = xc - hx; det[1] = yc - hy;
    det[2] = xc + hx; det[3] = yc + hy;
    det[4] = score;

    float* kout = out + (size_t)BATCH * TOPK * 5 + (size_t)t * 51;
    #pragma unroll
    for (int p = 0; p < 17; ++p) {
        kout[p * 3 + 0] = kp[(size_t)(2 * p)     * hw] * fs + px;
        kout[p * 3 + 1] = kp[(size_t)(2 * p + 1) * hw] * fs + py;
        kout[p * 3 + 2] = sigmoidf(vp[(size_t)p * hw]);
    }
}

// ---------------------------------------------------------------------------
// Launch. Input order: cls0-2, bbox0-2, obj0-2, kpt0-2, vis0-2.
// Workspace: keep (B*TOPK i32) = 128 KB.
// ---------------------------------------------------------------------------
extern "C" void kernel_launch(void* const* d_in, const int* in_sizes, int n_in,
                              void* d_out, int out_size, void* d_ws, size_t ws_size,
                              hipStream_t stream) {
    const float* cls0 = (const float*)d_in[0];
    const float* cls1 = (const float*)d_in[1];
    const float* cls2 = (const float*)d_in[2];
    const float* bb0  = (const float*)d_in[3];
    const float* bb1  = (const float*)d_in[4];
    const float* bb2  = (const float*)d_in[5];
    const float* ob0  = (const float*)d_in[6];
    const float* ob1  = (const float*)d_in[7];
    const float* ob2  = (const float*)d_in[8];
    const float* kp0  = (const float*)d_in[9];
    const float* kp1  = (const float*)d_in[10];
    const float* kp2  = (const float*)d_in[11];
    const float* vs0  = (const float*)d_in[12];
    const float* vs1  = (const float*)d_in[13];
    const float* vs2  = (const float*)d_in[14];

    int*   keep = (int*)d_ws;
    float* outf = (float*)d_out;

    topk_fused_kernel<<<BATCH, TPB, LDS_TOTAL, stream>>>(
        cls0, cls1, cls2, ob0, ob1, ob2, keep);

    const int total = BATCH * TOPK;
    gather_decode_kernel<<<(total + 127) / 128, 128, 0, stream>>>(
        cls0, cls1, cls2, bb0, bb1, bb2, ob0, ob1, ob2,
        kp0, kp1, kp2, vs0, vs1, vs2, keep, outf);
}